// Decoder_79121887527344
// MI455X (gfx1250) — compile-verified
//
#include <hip/hip_runtime.h>
#include <math.h>

#define V_SIZE 50257
#define H_SIZE 1024
#define KB 32          // K-block staged per buffer
#define KP (KB + 4)    // padded LDS row stride (conflict-free ds_load_b64)

typedef __attribute__((ext_vector_type(2))) float v2f;
typedef __attribute__((ext_vector_type(8))) float v8f;

// Workspace layout (float indices)
#define WS_X     0                       // relu(emb[token])        [1024]
#define WS_HNEW  1024                    // h_new                   [1024]
#define WS_GI    2048                    // w_ih@x + b_ih           [3072]
#define WS_GH    (2048 + 3072)           // w_hh@h + b_hh           [3072]
#define WS_LOG   8192                    // logits                  [50257]
#define WS_REDM  (8192 + 50257)          // ordered-uint running max [1]
#define WS_REDS  (8192 + 50258)          // float running sum        [1]

// Monotonic float<->uint mapping so atomicMax(unsigned) orders floats.
__device__ __forceinline__ unsigned ord_f32(float f) {
  unsigned u = __float_as_uint(f);
  return (u & 0x80000000u) ? ~u : (u | 0x80000000u);
}
__device__ __forceinline__ float deord_f32(unsigned o) {
  return (o & 0x80000000u) ? __uint_as_float(o & 0x7FFFFFFFu)
                           : __uint_as_float(~o);
}

// ---------------------------------------------------------------------------
// Kernel 1: x = relu(emb[token]); init softmax reduction cells.
// ---------------------------------------------------------------------------
__global__ void prep_kernel(const int* __restrict__ token,
                            const float* __restrict__ emb,
                            float* __restrict__ ws) {
  const int t = token[0];
  const float* row = emb + (size_t)t * H_SIZE;
  for (int i = threadIdx.x; i < H_SIZE; i += blockDim.x)
    ws[WS_X + i] = fmaxf(row[i], 0.0f);
  if (threadIdx.x == 0) {
    ((unsigned*)ws)[WS_REDM] = 0u;   // ord() of any real float exceeds 0
    ws[WS_REDS] = 0.0f;
  }
}

// ---------------------------------------------------------------------------
// WMMA f32 GEMV: out[M] = W[M x K] @ vec[K] + bias[M].
// Block = 256 threads = 8 wave32; each wave owns 16 rows.
// Data movement: double-buffered GLOBAL_LOAD_ASYNC_TO_LDS_B128 (ASYNCcnt),
// 4 copy instructions per 16x32 f32 tile, 128 contiguous bytes per row.
// Compute: V_WMMA_F32_16X16X4_F32 with the vector in column N=0 of B
// (bandwidth-bound, unused columns are free; f32 keeps reference numerics).
// ---------------------------------------------------------------------------
__global__ void __launch_bounds__(256)
gemv_wmma_kernel(const float* __restrict__ W, const float* __restrict__ vec,
                 const float* __restrict__ bias, float* __restrict__ out,
                 int M, int K) {
  __shared__ float lds[8][2][16][KP];    // wave-private double buffers (36 KB)
  const int lane = threadIdx.x & 31;
  const int wave = threadIdx.x >> 5;
  const int half = lane >> 4;            // 0: A K-offsets {0,1}; 1: {2,3}
  const int l15  = lane & 15;
  const int row0 = blockIdx.x * 128 + wave * 16;

  // Staging coordinates: 8 lanes per row -> 4 rows per copy instruction.
  const int srow = lane >> 3;            // 0..3
  const int scol = (lane & 7) * 4;       // 0..28

  const unsigned long long Wbase = (unsigned long long)(size_t)W;

  // Issue one K-block (16 rows x KB cols) as 4 async b128 copies to LDS.
  auto stage = [&](int buf, int kb) {
#pragma unroll
    for (int i = 0; i < 4; ++i) {
      int r  = i * 4 + srow;
      int rg = row0 + r; if (rg > M - 1) rg = M - 1;   // clamp: EXEC stays full
      unsigned goff  = ((unsigned)rg * (unsigned)K + (unsigned)(kb + scol)) * 4u;
      // LDS aperture keeps the byte offset in the low 32 bits.
      unsigned laddr = (unsigned)(size_t)&lds[wave][buf][r][scol];
      asm volatile("global_load_async_to_lds_b128 %0, %1, %2"
                   :: "v"(laddr), "v"(goff), "s"(Wbase) : "memory");
    }
  };

  v8f c = {};
  stage(0, 0);
  const int NB = K / KB;
  for (int t = 0; t < NB; ++t) {
    const int kb  = t * KB;
    const int buf = t & 1;
    if (t + 1 < NB) {
      stage(buf ^ 1, kb + KB);          // overlap next tile with this compute
      // Loads complete in order: <=4 outstanding => previous tile landed.
      asm volatile("s_wait_asynccnt 0x4" ::: "memory");
    } else {
      asm volatile("s_wait_asynccnt 0x0" ::: "memory");
    }
#pragma unroll
    for (int kk = 0; kk < KB; kk += 4) {
      // A fragment (16x4): lanes 0-15 hold (M=lane, K={0,1}),
      //                    lanes 16-31 hold (M=lane-16, K={2,3}).
      v2f a = *(const v2f*)&lds[wave][buf][l15][kk + half * 2];
      // B fragment (4x16): only column N=0 (lanes 0 and 16) carries vec.
      v2f b = {0.0f, 0.0f};
      if (l15 == 0) {
        const float* vp = vec + kb + kk + half * 2;
        b.x = vp[0]; b.y = vp[1];
      }
      c = __builtin_amdgcn_wmma_f32_16x16x4_f32(false, a, false, b,
                                                (short)0, c, false, false);
    }
  }
  // D layout: VGPR j -> (M=j, N=lane) lanes 0-15 ; (M=j+8, N=lane-16) 16-31.
  // Column N=0 lives in lanes 0 (rows 0-7) and 16 (rows 8-15).
  if (l15 == 0) {
#pragma unroll
    for (int m = 0; m < 8; ++m) {
      int row = row0 + half * 8 + m;
      if (row < M) out[row] = c[m] + bias[row];
    }
  }
}

// ---------------------------------------------------------------------------
// Kernel 3: GRU gate math -> h_new (into ws and d_out[V:V+H]).
// ---------------------------------------------------------------------------
__global__ void gru_gate_kernel(const float* __restrict__ gi,
                                const float* __restrict__ gh,
                                const float* __restrict__ h,
                                float* __restrict__ hnew_ws,
                                float* __restrict__ d_out) {
  for (int j = threadIdx.x; j < H_SIZE; j += blockDim.x) {
    float r  = 1.0f / (1.0f + __expf(-(gi[j] + gh[j])));
    float z  = 1.0f / (1.0f + __expf(-(gi[H_SIZE + j] + gh[H_SIZE + j])));
    float n  = tanhf(gi[2 * H_SIZE + j] + r * gh[2 * H_SIZE + j]);
    float hn = (1.0f - z) * n + z * h[j];
    hnew_ws[j] = hn;
    d_out[V_SIZE + j] = hn;
  }
}

// ---------------------------------------------------------------------------
// Softmax pass 1: grid-wide max via block reduce + atomicMax(ordered uint).
// ---------------------------------------------------------------------------
__global__ void softmax_max_kernel(const float* __restrict__ logits,
                                   unsigned* __restrict__ redm) {
  __shared__ float sm[256];
  float m = -INFINITY;
  for (int i = blockIdx.x * blockDim.x + threadIdx.x; i < V_SIZE;
       i += gridDim.x * blockDim.x)
    m = fmaxf(m, logits[i]);
  sm[threadIdx.x] = m;
  __syncthreads();
  for (int s = 128; s > 0; s >>= 1) {
    if (threadIdx.x < s) sm[threadIdx.x] = fmaxf(sm[threadIdx.x], sm[threadIdx.x + s]);
    __syncthreads();
  }
  if (threadIdx.x == 0) atomicMax(redm, ord_f32(sm[0]));
}

// Softmax pass 2: e = exp(l - max) -> d_out; block sum -> atomicAdd.
__global__ void softmax_expsum_kernel(const float* __restrict__ logits,
                                      const unsigned* __restrict__ redm,
                                      float* __restrict__ reds,
                                      float* __restrict__ d_out) {
  __shared__ float sm[256];
  const float mx = deord_f32(*redm);
  float s = 0.0f;
  for (int i = blockIdx.x * blockDim.x + threadIdx.x; i < V_SIZE;
       i += gridDim.x * blockDim.x) {
    float e = __expf(logits[i] - mx);
    d_out[i] = e;
    s += e;
  }
  sm[threadIdx.x] = s;
  __syncthreads();
  for (int st = 128; st > 0; st >>= 1) {
    if (threadIdx.x < st) sm[threadIdx.x] += sm[threadIdx.x + st];
    __syncthreads();
  }
  if (threadIdx.x == 0) atomicAdd(reds, sm[0]);
}

// Softmax pass 3: normalize in place.
__global__ void softmax_norm_kernel(float* __restrict__ d_out,
                                    const float* __restrict__ reds) {
  const float inv = 1.0f / *reds;
  for (int i = blockIdx.x * blockDim.x + threadIdx.x; i < V_SIZE;
       i += gridDim.x * blockDim.x)
    d_out[i] *= inv;
}

// ---------------------------------------------------------------------------
extern "C" void kernel_launch(void* const* d_in, const int* in_sizes, int n_in,
                              void* d_out, int out_size, void* d_ws, size_t ws_size,
                              hipStream_t stream) {
  (void)in_sizes; (void)n_in; (void)out_size; (void)ws_size;
  const int*   token = (const int*)  d_in[0];
  const float* h     = (const float*)d_in[1];
  const float* emb   = (const float*)d_in[2];
  const float* w_ih  = (const float*)d_in[3];
  const float* w_hh  = (const float*)d_in[4];
  const float* b_ih  = (const float*)d_in[5];
  const float* b_hh  = (const float*)d_in[6];
  const float* w_out = (const float*)d_in[7];
  const float* b_out = (const float*)d_in[8];
  float* out = (float*)d_out;
  float* ws  = (float*)d_ws;

  prep_kernel<<<1, 256, 0, stream>>>(token, emb, ws);
  gemv_wmma_kernel<<<24, 256, 0, stream>>>(w_ih, ws + WS_X, b_ih,
                                           ws + WS_GI, 3 * H_SIZE, H_SIZE);
  gemv_wmma_kernel<<<24, 256, 0, stream>>>(w_hh, h, b_hh,
                                           ws + WS_GH, 3 * H_SIZE, H_SIZE);
  gru_gate_kernel<<<1, 256, 0, stream>>>(ws + WS_GI, ws + WS_GH, h,
                                         ws + WS_HNEW, out);
  gemv_wmma_kernel<<<(V_SIZE + 127) / 128, 256, 0, stream>>>(
      w_out, ws + WS_HNEW, b_out, ws + WS_LOG, V_SIZE, H_SIZE);
  softmax_max_kernel<<<64, 256, 0, stream>>>(ws + WS_LOG,
                                             (unsigned*)(ws + WS_REDM));
  softmax_expsum_kernel<<<64, 256, 0, stream>>>(ws + WS_LOG,
                                                (const unsigned*)(ws + WS_REDM),
                                                ws + WS_REDS, out);
  softmax_norm_kernel<<<64, 256, 0, stream>>>(out, ws + WS_REDS);
}